// PlanarFlow_4999341932932
// MI455X (gfx1250) — compile-verified
//
#include <hip/hip_runtime.h>
#include <stdint.h>

#define BATCH   16384
#define STEPS   32
#define LAT     64
#define IN_W    (STEPS * (2 * LAT + 1) + 2 * LAT)   // 4256
#define DEPTH   8                                    // async prefetch ring depth (stages)
#define WAVES_PER_BLOCK 8
#define ROWS_PER_WAVE   2

// ---------------------------------------------------------------------------
// CDNA5 async global->LDS copy (ASYNCcnt path, CDNA5 ISA)
// VDST = VGPR holding LDS byte address, VADDR = 64-bit global address (GV mode).
// ---------------------------------------------------------------------------
__device__ __forceinline__ void async_copy_b128(uint32_t lds_byte_addr, const float* gaddr) {
    asm volatile("global_load_async_to_lds_b128 %0, %1, off"
                 :: "v"(lds_byte_addr), "v"(gaddr)
                 : "memory");
}
template <int N>
__device__ __forceinline__ void wait_async_le() {
    asm volatile("s_wait_asynccnt %0" :: "n"(N) : "memory");
}
__device__ __forceinline__ void wait_ds0() {
    asm volatile("s_wait_dscnt 0x0" ::: "memory");
}

// Butterfly sum across a 16-lane half-wave (masks stay inside the half).
__device__ __forceinline__ float red16(float v) {
    v += __shfl_xor(v, 1);
    v += __shfl_xor(v, 2);
    v += __shfl_xor(v, 4);
    v += __shfl_xor(v, 8);
    return v;
}

struct Ctx {
    const float* rowp;    // this lane's row base in global memory
    const float* slab;    // generic ptr to this thread's LDS lane slot (stage0, mat u)
    uint32_t slab_lds;    // LDS byte address of the same slot
    float2 bp;            // biases b[2k], b[2k+1] held by lane lsub=k
    int lane;
    int lsub;
};

// Issue async copies of stage `stage`'s u/w data into ring slot `slot`.
__device__ __forceinline__ void issue_stage(const Ctx& c, int stage, int slot) {
    const float* gu = c.rowp + 2 * LAT + stage * LAT + c.lsub * 4;
    const uint32_t la = c.slab_lds + (uint32_t)slot * 1024u;   // 2 mats * 128 fl * 4B
    async_copy_b128(la,        gu);                  // u
    async_copy_b128(la + 512u, gu + STEPS * LAT);    // w
}

// One planar-flow step consuming ring slot `st`.
__device__ __forceinline__ void flow_step(int s, int st, const Ctx& c,
                                          float4& z, float& acc) {
    const float4 uu = *(const float4*)(c.slab + st * 256);
    const float4 ww = *(const float4*)(c.slab + st * 256 + 128);

    float hp = ww.x * z.x + ww.y * z.y + ww.z * z.z + ww.w * z.w;
    float up = uu.x * ww.x + uu.y * ww.y + uu.z * ww.z + uu.w * ww.w;
    hp = red16(hp);
    up = red16(up);

    const float bv = (s & 1) ? c.bp.y : c.bp.x;
    const float bs = __shfl(bv, (c.lane & 16) | (s >> 1));

    const float t = tanhf(hp + bs);
    acc -= logf(fabsf((1.0f - t * t) * up + 1.0f));

    z.x += uu.x * t;
    z.y += uu.y * t;
    z.z += uu.z * t;
    z.w += uu.w * t;
}

// Tail steps with exact constant async-wait counts (in-order completion).
template <int S>
__device__ __forceinline__ void tail_steps(const Ctx& c, float4& z, float& acc) {
    if constexpr (S < STEPS) {
        wait_async_le<2 * (STEPS - 1 - S)>();
        flow_step(S, S & (DEPTH - 1), c, z, acc);
        tail_steps<S + 1>(c, z, acc);
    }
}

__global__ __launch_bounds__(256)
void planar_flow_kernel(const float* __restrict__ inp,
                        const float* __restrict__ noise,
                        float* __restrict__ z_out,
                        float* __restrict__ row_loss) {
    // LDS ring: [wave][stage][u|w][2 rows * 64 floats] = 64 KB / block
    __shared__ float ldsbuf[WAVES_PER_BLOCK][DEPTH][2][ROWS_PER_WAVE * LAT];

    const int tid   = threadIdx.x;
    const int wv    = tid >> 5;          // wave within block
    const int lane  = tid & 31;
    const int lsub  = lane & 15;         // column group within row (4 cols each)
    const int rhalf = lane >> 4;         // 0 -> row A, 1 -> row B
    const int row   = (blockIdx.x * WAVES_PER_BLOCK + wv) * ROWS_PER_WAVE + rhalf;

    Ctx c;
    c.rowp = inp + (size_t)row * IN_W;
    c.lane = lane;
    c.lsub = lsub;
    const int lelem = lane * 4;          // == rhalf*64 + lsub*4
    c.slab     = &ldsbuf[wv][0][0][lelem];
    c.slab_lds = (uint32_t)(uintptr_t)c.slab;
    c.bp = *(const float2*)(c.rowp + (2 + 2 * STEPS) * LAT + lsub * 2);

    // ---- z0 = mu + noise * exp(0.5 * logsigma2); KL partials -------------
    const float4 mu = *(const float4*)(c.rowp + lsub * 4);
    const float4 ls = *(const float4*)(c.rowp + LAT + lsub * 4);
    const float4 nz = *(const float4*)(noise + (size_t)row * LAT + lsub * 4);

    float4 z;
    z.x = mu.x + nz.x * expf(0.5f * ls.x);
    z.y = mu.y + nz.y * expf(0.5f * ls.y);
    z.z = mu.z + nz.z * expf(0.5f * ls.z);
    z.w = mu.w + nz.w * expf(0.5f * ls.w);

    float klp = nz.x * nz.x + nz.y * nz.y + nz.z * nz.z + nz.w * nz.w
              + ls.x + ls.y + ls.z + ls.w;
    float acc = -0.5f * red16(klp);      // row-local loss accumulator

    // ---- prologue: fill the async ring (DEPTH stages * 2 copies) ----------
#pragma unroll
    for (int s = 0; s < DEPTH; ++s) issue_stage(c, s, s);

    // ---- steady state: consume stage s, refill slot with stage s+DEPTH ----
    for (int s = 0; s < STEPS - DEPTH; ++s) {
        // issued through stage s+DEPTH-1 => cnt<=2*(DEPTH-1) means stage s landed
        wait_async_le<2 * (DEPTH - 1)>();
        const int st = s & (DEPTH - 1);
        flow_step(s, st, c, z, acc);
        wait_ds0();                       // WAR: LDS reads done before overwrite
        issue_stage(c, s + DEPTH, st);
    }

    // ---- tail: exact waits, no refills -----------------------------------
    tail_steps<STEPS - DEPTH>(c, z, acc);

    // ---- outputs ----------------------------------------------------------
    *(float4*)(z_out + (size_t)row * LAT + lsub * 4) = z;
    if (lsub == 0) row_loss[row] = acc;
}

// Deterministic fixed-order reduction of the 16384 per-row loss terms.
__global__ __launch_bounds__(256)
void reduce_loss_kernel(const float* __restrict__ row_loss, float* __restrict__ out_scalar) {
    __shared__ float smem[256];
    float a = 0.0f;
    for (int i = threadIdx.x; i < BATCH; i += 256) a += row_loss[i];
    smem[threadIdx.x] = a;
    __syncthreads();
    for (int off = 128; off > 0; off >>= 1) {
        if (threadIdx.x < off) smem[threadIdx.x] += smem[threadIdx.x + off];
        __syncthreads();
    }
    if (threadIdx.x == 0) *out_scalar = smem[0] * (1.0f / (float)BATCH);
}

extern "C" void kernel_launch(void* const* d_in, const int* in_sizes, int n_in,
                              void* d_out, int out_size, void* d_ws, size_t ws_size,
                              hipStream_t stream) {
    const float* inp   = (const float*)d_in[0];   // (BATCH, IN_W) f32
    const float* noise = (const float*)d_in[1];   // (BATCH, LAT)  f32
    float* out = (float*)d_out;                   // z (BATCH*LAT) then scalar loss
    float* ws  = (float*)d_ws;                    // BATCH floats of per-row loss

    const int rows_per_block = WAVES_PER_BLOCK * ROWS_PER_WAVE;   // 16
    planar_flow_kernel<<<BATCH / rows_per_block, 256, 0, stream>>>(inp, noise, out, ws);
    reduce_loss_kernel<<<1, 256, 0, stream>>>(ws, out + (size_t)BATCH * LAT);
}